// MILoss_77781857731276
// MI455X (gfx1250) — compile-verified
//
#include <hip/hip_runtime.h>
#include <math.h>

#define NBINS 30
#define NB2   (NBINS * NBINS)      // 900
#define DCOLS 16
#define EPSF  1e-10f

// ---- workspace word offsets (ws is uint32 words) ----
#define WS_L1     0
#define WS_YPMIN  1
#define WS_YPMAX  2
#define WS_UMIN   3     // 16 words
#define WS_UMAX   19    // 16 words
#define WS_SMIN   35    // 16 words
#define WS_SMAX   51    // 16 words
#define WS_MI2D   67
#define WS_MIUS   68
#define WS_HIST   128                    // 32 * 900 uints
#define WS_WORDS  (WS_HIST + 32 * NB2)

typedef __attribute__((ext_vector_type(2))) float v2f;
typedef __attribute__((ext_vector_type(8))) float v8f;

#if defined(__AMDGCN__) && __has_builtin(__builtin_amdgcn_wmma_f32_16x16x4_f32)
#define HAVE_WMMA 1
#else
#define HAVE_WMMA 0
#endif

// order-preserving float<->uint encoding so unsigned atomicMin/Max give float min/max
__device__ __forceinline__ unsigned encf(float f) {
    unsigned u = __float_as_uint(f);
    return (u & 0x80000000u) ? ~u : (u | 0x80000000u);
}
__device__ __forceinline__ float decf(unsigned u) {
    unsigned b = (u & 0x80000000u) ? (u & 0x7FFFFFFFu) : ~u;
    return __uint_as_float(b);
}

// ---------------- init workspace (harness poisons ws) ----------------
__global__ void k_init(unsigned* __restrict__ ws) {
    int idx = blockIdx.x * blockDim.x + threadIdx.x;
    int stride = gridDim.x * blockDim.x;
    for (int i = idx; i < WS_WORDS; i += stride) {
        bool is_min = (i == WS_YPMIN) ||
                      (i >= WS_UMIN && i < WS_UMIN + 16) ||
                      (i >= WS_SMIN && i < WS_SMIN + 16);
        ws[i] = is_min ? 0xFFFFFFFFu : 0u;
    }
}

// ------------- pass 1: per-column min/max + L1 sum (single stream of all inputs) -------------
__global__ void k_minmax_l1(const float* __restrict__ yt, const float* __restrict__ yp,
                            const float* __restrict__ u,  const float* __restrict__ s,
                            unsigned* __restrict__ ws, int n) {
    int tid = blockIdx.x * blockDim.x + threadIdx.x;
    int stride = gridDim.x * blockDim.x;
    float l1 = 0.f;
    float ymn = INFINITY, ymx = -INFINITY;
    float umn[16], umx[16], smn[16], smx[16];
#pragma unroll
    for (int c = 0; c < 16; ++c) { umn[c] = INFINITY; umx[c] = -INFINITY; smn[c] = INFINITY; smx[c] = -INFINITY; }

    for (int r = tid; r < n; r += stride) {
        float p = yp[r];
        l1 += fabsf(yt[r] - p);
        ymn = fminf(ymn, p); ymx = fmaxf(ymx, p);
        const float4* u4 = (const float4*)(u + (size_t)r * 16);
        const float4* s4 = (const float4*)(s + (size_t)r * 16);
#pragma unroll
        for (int q = 0; q < 4; ++q) {
            float4 a = u4[q], b = s4[q];
            umn[4*q+0] = fminf(umn[4*q+0], a.x); umx[4*q+0] = fmaxf(umx[4*q+0], a.x);
            umn[4*q+1] = fminf(umn[4*q+1], a.y); umx[4*q+1] = fmaxf(umx[4*q+1], a.y);
            umn[4*q+2] = fminf(umn[4*q+2], a.z); umx[4*q+2] = fmaxf(umx[4*q+2], a.z);
            umn[4*q+3] = fminf(umn[4*q+3], a.w); umx[4*q+3] = fmaxf(umx[4*q+3], a.w);
            smn[4*q+0] = fminf(smn[4*q+0], b.x); smx[4*q+0] = fmaxf(smx[4*q+0], b.x);
            smn[4*q+1] = fminf(smn[4*q+1], b.y); smx[4*q+1] = fmaxf(smx[4*q+1], b.y);
            smn[4*q+2] = fminf(smn[4*q+2], b.z); smx[4*q+2] = fmaxf(smx[4*q+2], b.z);
            smn[4*q+3] = fminf(smn[4*q+3], b.w); smx[4*q+3] = fmaxf(smx[4*q+3], b.w);
        }
    }

    __shared__ float    shl1[256];
    __shared__ unsigned shmn[33], shmx[33];
    int t = threadIdx.x;
    if (t < 33) { shmn[t] = 0xFFFFFFFFu; shmx[t] = 0u; }
    shl1[t] = l1;
    __syncthreads();
    atomicMin(&shmn[0], encf(ymn)); atomicMax(&shmx[0], encf(ymx));
#pragma unroll
    for (int c = 0; c < 16; ++c) {
        atomicMin(&shmn[1 + c],  encf(umn[c])); atomicMax(&shmx[1 + c],  encf(umx[c]));
        atomicMin(&shmn[17 + c], encf(smn[c])); atomicMax(&shmx[17 + c], encf(smx[c]));
    }
    __syncthreads();
    for (int off = 128; off > 0; off >>= 1) { if (t < off) shl1[t] += shl1[t + off]; __syncthreads(); }
    if (t == 0) {
        atomicAdd((float*)&ws[WS_L1], shl1[0]);
        atomicMin(&ws[WS_YPMIN], shmn[0]); atomicMax(&ws[WS_YPMAX], shmx[0]);
    }
    if (t >= 1 && t < 17)  { atomicMin(&ws[WS_UMIN + t - 1],  shmn[t]); atomicMax(&ws[WS_UMAX + t - 1],  shmx[t]); }
    if (t >= 17 && t < 33) { atomicMin(&ws[WS_SMIN + t - 17], shmn[t]); atomicMax(&ws[WS_SMAX + t - 17], shmx[t]); }
}

// ------------- pass 2: 16 private 30x30 histograms in LDS, one flush per workgroup -------------
__global__ void k_hist(const float* __restrict__ X, const float* __restrict__ yp,
                       const unsigned* __restrict__ ws, unsigned* __restrict__ ghist,
                       int minoff, int maxoff, int n) {
    __shared__ unsigned lh[16 * NB2];   // 57.6 KB (WGP LDS is 320 KB)
    int t = threadIdx.x;
    for (int i = t; i < 16 * NB2; i += blockDim.x) lh[i] = 0u;

    float ypmin = decf(ws[WS_YPMIN]);
    float ypmax = decf(ws[WS_YPMAX]);
    float ysc = (ypmax > ypmin) ? (float)NBINS / (ypmax - ypmin) : 0.f;
    float cmin[16], csc[16];
#pragma unroll
    for (int c = 0; c < 16; ++c) {
        float mn = decf(ws[minoff + c]), mx = decf(ws[maxoff + c]);
        cmin[c] = mn;
        csc[c] = (mx > mn) ? (float)NBINS / (mx - mn) : 0.f;
    }
    __syncthreads();

    int tid = blockIdx.x * blockDim.x + t;
    int stride = gridDim.x * blockDim.x;
    for (int r = tid; r < n; r += stride) {
        float p = yp[r];
        int iy = (int)((p - ypmin) * ysc);
        iy = iy < 0 ? 0 : (iy > NBINS - 1 ? NBINS - 1 : iy);
        const float4* x4 = (const float4*)(X + (size_t)r * 16);
        float v[16];
#pragma unroll
        for (int q = 0; q < 4; ++q) {
            float4 a = x4[q];
            v[4*q] = a.x; v[4*q+1] = a.y; v[4*q+2] = a.z; v[4*q+3] = a.w;
        }
#pragma unroll
        for (int c = 0; c < 16; ++c) {
            int ix = (int)((v[c] - cmin[c]) * csc[c]);
            ix = ix < 0 ? 0 : (ix > NBINS - 1 ? NBINS - 1 : ix);
            atomicAdd(&lh[c * NB2 + ix * NBINS + iy], 1u);
        }
    }
    __syncthreads();
    for (int i = t; i < 16 * NB2; i += blockDim.x) {
        unsigned cv = lh[i];
        if (cv) atomicAdd(&ghist[i], cv);
    }
}

// ------------- pass 3: MI of each 30x30 table; marginals via V_WMMA_F32_16X16X4_F32 -------------
__global__ void __launch_bounds__(32) k_mi2d(const unsigned* __restrict__ ghist,
                                             float* __restrict__ acc_out) {
    __shared__ float H[32][33];
    __shared__ float rsum[32], csum[32];
    __shared__ float tot_s;
    __shared__ float part[32];
    int lane = threadIdx.x;
    const unsigned* h = ghist + (size_t)blockIdx.x * NB2;

    for (int i = lane; i < 32 * 32; i += 32) H[i >> 5][i & 31] = 0.f;
    __syncthreads();
    for (int i = lane; i < NB2; i += 32) H[i / NBINS][i % NBINS] = (float)h[i];
    __syncthreads();

#if HAVE_WMMA
    {
        // A-matrix 16x4 f32 layout: lanes 0-15 M=0..15 hold K=0,1; lanes 16-31 hold K=2,3
        int mrow  = (lane & 15);
        int khalf = (lane >> 4) * 2;
        v2f bones; bones.x = 1.f; bones.y = 1.f;   // B = 4x16 all-ones -> D[m][*] = rowsum(A_m)
#pragma unroll
        for (int blk = 0; blk < 2; ++blk) {        // row sums (rows blk*16..blk*16+15)
            v8f acc = {0.f,0.f,0.f,0.f,0.f,0.f,0.f,0.f};
#pragma unroll
            for (int kb = 0; kb < 8; ++kb) {
                v2f a;
                a.x = H[blk * 16 + mrow][kb * 4 + khalf + 0];
                a.y = H[blk * 16 + mrow][kb * 4 + khalf + 1];
                acc = __builtin_amdgcn_wmma_f32_16x16x4_f32(false, a, false, bones,
                                                            (short)0, acc, false, false);
            }
            if (lane == 0 || lane == 16) {         // C/D: VGPR v -> M=v (lanes 0-15) / M=v+8 (16-31)
                int mb = blk * 16 + (lane >> 4) * 8;
#pragma unroll
                for (int v = 0; v < 8; ++v) rsum[mb + v] = acc[v];
            }
        }
#pragma unroll
        for (int blk = 0; blk < 2; ++blk) {        // col sums = row sums of H^T
            v8f acc = {0.f,0.f,0.f,0.f,0.f,0.f,0.f,0.f};
#pragma unroll
            for (int kb = 0; kb < 8; ++kb) {
                v2f a;
                a.x = H[kb * 4 + khalf + 0][blk * 16 + mrow];
                a.y = H[kb * 4 + khalf + 1][blk * 16 + mrow];
                acc = __builtin_amdgcn_wmma_f32_16x16x4_f32(false, a, false, bones,
                                                            (short)0, acc, false, false);
            }
            if (lane == 0 || lane == 16) {
                int mb = blk * 16 + (lane >> 4) * 8;
#pragma unroll
                for (int v = 0; v < 8; ++v) csum[mb + v] = acc[v];
            }
        }
    }
#else
    {
        float rs = 0.f, cs = 0.f;
        for (int j = 0; j < 32; ++j) { rs += H[lane][j]; cs += H[j][lane]; }
        rsum[lane] = rs; csum[lane] = cs;
    }
#endif
    __syncthreads();
    if (lane == 0) {
        float ts = 0.f;
        for (int i = 0; i < NBINS; ++i) ts += rsum[i];
        tot_s = ts;
    }
    __syncthreads();
    float inv = (tot_s > 0.f) ? 1.f / tot_s : 0.f;
    float mi = 0.f;
    for (int i = lane; i < NB2; i += 32) {
        int bi = i / NBINS, bj = i % NBINS;
        float px  = rsum[bi] * inv; if (px  == 0.f) px  = EPSF;
        float py  = csum[bj] * inv; if (py  == 0.f) py  = EPSF;
        float pxy = H[bi][bj] * inv; if (pxy == 0.f) pxy = EPSF;
        mi += pxy * logf(pxy / (px * py));
    }
    part[lane] = mi;
    __syncthreads();
    if (lane == 0) {
        float sum = 0.f;
        for (int i = 0; i < 32; ++i) sum += part[i];
        atomicAdd(acc_out, sum);
    }
}

// ------------- pass 4: mi_us over first 16 rows (16 elements each) -------------
__global__ void k_mius(const float* __restrict__ u, const float* __restrict__ s,
                       float* __restrict__ out) {
    int lane = threadIdx.x;
    if (lane >= DCOLS) return;
    float uv[16], sv[16];
#pragma unroll
    for (int c = 0; c < 16; ++c) { uv[c] = u[lane * 16 + c]; sv[c] = s[lane * 16 + c]; }
    float umn = uv[0], umx = uv[0], smn = sv[0], smx = sv[0];
#pragma unroll
    for (int c = 1; c < 16; ++c) {
        umn = fminf(umn, uv[c]); umx = fmaxf(umx, uv[c]);
        smn = fminf(smn, sv[c]); smx = fmaxf(smx, sv[c]);
    }
    float ust = (umx - umn) / (float)(NBINS - 1);   // 30 edges: linspace(min,max,30)
    float sst = (smx - smn) / (float)(NBINS - 1);
    int lu[16], ls[16];
    for (int c = 0; c < 16; ++c) {
        int cu = 0, cs = 0;
        for (int e = 0; e < NBINS; ++e) {           // searchsorted 'right' = #edges <= v
            cu += (umn + ust * (float)e) <= uv[c];
            cs += (smn + sst * (float)e) <= sv[c];
        }
        lu[c] = cu; ls[c] = cs;
    }
    const float nf = 16.f;
    float mi = 0.f;
    for (int i = 0; i < 16; ++i) {
        bool first = true;
        for (int j = 0; j < i; ++j)
            if (lu[j] == lu[i] && ls[j] == ls[i]) first = false;
        if (!first) continue;
        int cc = 0, aa = 0, bb = 0;
        for (int j = 0; j < 16; ++j) {
            aa += (lu[j] == lu[i]);
            bb += (ls[j] == ls[i]);
            cc += (lu[j] == lu[i] && ls[j] == ls[i]);
        }
        mi += ((float)cc / nf) * logf((float)cc * nf / ((float)aa * (float)bb));
    }
    atomicAdd(out, mi);
}

// ------------- pass 5: combine -------------
__global__ void k_final(const unsigned* __restrict__ ws, float* __restrict__ out, float invn) {
    float l1   = __uint_as_float(ws[WS_L1]);
    float mi2  = __uint_as_float(ws[WS_MI2D]);
    float mius = __uint_as_float(ws[WS_MIUS]);
    out[0] = l1 * invn + 0.1f * (mi2 - 0.5f * mius);
}

extern "C" void kernel_launch(void* const* d_in, const int* in_sizes, int n_in,
                              void* d_out, int out_size, void* d_ws, size_t ws_size,
                              hipStream_t stream) {
    const float* yt = (const float*)d_in[0];
    const float* yp = (const float*)d_in[1];
    const float* u  = (const float*)d_in[4];
    const float* s  = (const float*)d_in[5];
    unsigned* ws = (unsigned*)d_ws;
    float* out = (float*)d_out;
    int n = in_sizes[0];

    k_init<<<128, 256, 0, stream>>>(ws);
    k_minmax_l1<<<512, 256, 0, stream>>>(yt, yp, u, s, ws, n);
    k_hist<<<256, 256, 0, stream>>>(u, yp, ws, ws + WS_HIST,            WS_UMIN, WS_UMAX, n);
    k_hist<<<256, 256, 0, stream>>>(s, yp, ws, ws + WS_HIST + 16 * NB2, WS_SMIN, WS_SMAX, n);
    k_mi2d<<<32, 32, 0, stream>>>(ws + WS_HIST, (float*)&ws[WS_MI2D]);
    k_mius<<<1, 32, 0, stream>>>(u, s, (float*)&ws[WS_MIUS]);
    k_final<<<1, 1, 0, stream>>>(ws, out, 1.f / (float)n);
}